// GraphConv_83030307766284
// MI455X (gfx1250) — compile-verified
//
#include <hip/hip_runtime.h>

#define NN     8192
#define ICH    128
#define FF     32
#define Y2NF   (2*NN*FF)          // 524288  (stride of channel axis in Y)
#define YSIZE  (2*2*NN*FF)        // 1048576 floats
#define ESIZE  ((size_t)NN*NN*2)  // 134217728 floats per edge tensor
#define KCH    8                  // k-split (deterministic partial buffers)
#define KPER   (NN/KCH)           // 1024 k per block
#define KB     32                 // k per LDS chunk
#define APAD   68                 // A-tile row stride in floats (32*2 + 4 pad -> conflict free)
#define CTPAD  36                 // CT-tile row stride in floats (32 + 4 pad  -> conflict free)
#define CT_FLOATS (FF*NN)         // 262144 floats (1 MB) in workspace

typedef float v2f __attribute__((ext_vector_type(2)));
typedef float v8f __attribute__((ext_vector_type(8)));

// ---- CDNA5 async LDS<->global DMA (ASYNCcnt-tracked, no VGPR transit) ----
__device__ __forceinline__ void async_load_b128(uint32_t lds_addr, const void* gaddr) {
    asm volatile("global_load_async_to_lds_b128 %0, %1, off"
                 :: "v"(lds_addr), "v"(gaddr) : "memory");
}
__device__ __forceinline__ void async_store_b128(void* gaddr, uint32_t lds_addr) {
    asm volatile("global_store_async_from_lds_b128 %0, %1, off"
                 :: "v"(gaddr), "v"(lds_addr) : "memory");
}
__device__ __forceinline__ void wait_async0() {
    asm volatile("s_wait_asynccnt 0" ::: "memory");
}
__device__ __forceinline__ uint32_t lds_off(const void* p) {
    return (uint32_t)(uintptr_t)p;   // low 32 bits of a shared pointer = LDS byte offset
}

// ---------------------------------------------------------------------------
// Kernel 1: CT[f][m] = alpha*W2[m][f] + (1-alpha)*( X[m,:]@W1[:,f] + b1[f] )
// ---------------------------------------------------------------------------
__global__ void prep_ct(const float* __restrict__ X, const float* __restrict__ W1,
                        const float* __restrict__ b1, const float* __restrict__ W2,
                        const float* __restrict__ alpha_p, float* __restrict__ CT) {
    __shared__ float sW1[ICH * FF];   // 16 KB
    __shared__ float sX[64 * ICH];    // 32 KB
    const int tid   = threadIdx.x;
    const int mbase = blockIdx.x * 64;

    const float4* W14  = (const float4*)W1;
    float4*       sW14 = (float4*)sW1;
#pragma unroll
    for (int j = 0; j < 4; ++j) sW14[tid + j * 256] = W14[tid + j * 256];

    const float4* X4  = (const float4*)(X + (size_t)mbase * ICH);
    float4*       sX4 = (float4*)sX;
#pragma unroll
    for (int j = 0; j < 8; ++j) sX4[tid + j * 256] = X4[tid + j * 256];
    __syncthreads();

    const float alpha = alpha_p[0];
    const int f  = tid & 31;
    const int mg = tid >> 5;          // 0..7
    const float bf = b1[f];

    for (int i = 0; i < 8; ++i) {
        const int ml = i * 8 + mg;    // local row 0..63
        float s = bf;
#pragma unroll 8
        for (int k = 0; k < ICH; ++k) s += sX[ml * ICH + k] * sW1[k * FF + f];
        const int m = mbase + ml;
        const float v = alpha * W2[(size_t)m * FF + f] + (1.0f - alpha) * s;
        CT[(size_t)f * NN + m] = v;
    }
}

// ---------------------------------------------------------------------------
// Kernel 2: fused  Ypart += E_tau @ C  (both channels, WMMA f32 16x16x4)
//           E staged HBM->LDS with global_load_async_to_lds_b128 and
//           identity-copied LDS->HBM with global_store_async_from_lds_b128.
// grid = (64 n-blocks, 8 k-chunks, 2 tensors), block = 256 (8 wave32s)
// ---------------------------------------------------------------------------
__global__ void gemm_main(const float* __restrict__ E0, const float* __restrict__ E1,
                          const float* __restrict__ CTg, float* __restrict__ d_out,
                          float* __restrict__ partials) {
    __shared__ float sCT[FF * CTPAD];          // 4.5 KB, shared by all waves
    __shared__ float sA[8 * 16 * APAD];        // 34 KB, 16x(KB m x 2ch) per wave

    const int tid  = threadIdx.x;
    const int lane = tid & 31;
    const int w    = tid >> 5;
    const int nblk = blockIdx.x;               // 0..63
    const int kc   = blockIdx.y;               // 0..7
    const int tau  = blockIdx.z;               // 0..1

    const float* Eg    = tau ? E1 : E0;
    float*       Ecopy = d_out + YSIZE + (size_t)tau * ESIZE;   // identity output slot
    const int rowg0 = nblk * 128 + w * 16;     // wave's first global row
    const int kbase = kc * KPER;

    const int h  = lane >> 4;                  // K half: 0 -> K=0,1 ; 1 -> K=2,3
    const int r  = lane & 15;                  // A row / B column within tile
    float* sAw = sA + w * (16 * APAD);

    // Per-lane staging geometry (constant across chunks)
    const int fr = tid >> 3;                   // CT row 0..31 (one float4/thread)
    const int mq = tid & 7;
    const uint32_t ctLds = lds_off(sCT + fr * CTPAD + mq * 4);
    uint32_t aLds[8];
    int      rrv[8], c4v[8];
#pragma unroll
    for (int i = 0; i < 8; ++i) {
        const int t = i * 32 + lane;           // 256 float4 per wave tile
        rrv[i] = t >> 4;                       // row 0..15 (16 float4 per row)
        c4v[i] = t & 15;
        aLds[i] = lds_off(sAw + rrv[i] * APAD + c4v[i] * 4);
    }

    v8f acc00 = {}, acc01 = {}, acc10 = {}, acc11 = {};

    for (int c = 0; c < KPER / KB; ++c) {      // 32 chunks of KB=32
        const int m0 = kbase + c * KB;
        __syncthreads();                       // all DS reads of prev chunk done
        wait_async0();                         // prev copy-stores drained (LDS reusable)

        // --- stage CT chunk (32 f x 32 m) + E tile (16 rows x 32 m x 2ch), async ---
        async_load_b128(ctLds, (const float4*)(CTg + (size_t)fr * NN + m0) + mq);
#pragma unroll
        for (int i = 0; i < 8; ++i) {
            const size_t g4 = (size_t)(rowg0 + rrv[i]) * (NN / 2) + (size_t)(m0 >> 1) + c4v[i];
            async_load_b128(aLds[i], (const float4*)Eg + g4);
        }
        wait_async0();                         // this wave's LDS tiles populated
        __syncthreads();                       // everyone's (incl. shared CT) populated

        // --- fused identity copy: LDS -> d_out tuple slot, async (overlaps WMMAs) ---
#pragma unroll
        for (int i = 0; i < 8; ++i) {
            const size_t g4 = (size_t)(rowg0 + rrv[i]) * (NN / 2) + (size_t)(m0 >> 1) + c4v[i];
            async_store_b128((float4*)Ecopy + g4, aLds[i]);
        }

        // --- 8 k-steps of 4, 4 WMMAs each (2 channels x 2 f-tiles) ---
#pragma unroll
        for (int ks = 0; ks < KB / 4; ++ks) {
            float4 av = *(const float4*)(sAw + r * APAD + ks * 8 + h * 4);
            v2f b0 = *(const v2f*)(sCT + r * CTPAD + ks * 4 + h * 2);
            v2f b1v = *(const v2f*)(sCT + (16 + r) * CTPAD + ks * 4 + h * 2);
            v2f a0 = {av.x, av.z};             // channel 0: K=h*2, h*2+1
            v2f a1 = {av.y, av.w};             // channel 1
            acc00 = __builtin_amdgcn_wmma_f32_16x16x4_f32(false, a0, false, b0,  (short)0, acc00, false, false);
            acc01 = __builtin_amdgcn_wmma_f32_16x16x4_f32(false, a0, false, b1v, (short)0, acc01, false, false);
            acc10 = __builtin_amdgcn_wmma_f32_16x16x4_f32(false, a1, false, b0,  (short)0, acc10, false, false);
            acc11 = __builtin_amdgcn_wmma_f32_16x16x4_f32(false, a1, false, b1v, (short)0, acc11, false, false);
        }
    }
    wait_async0();                             // drain final copy-stores

    // --- epilogue: scatter D per CDNA5 C/D layout (VGPR i: M=i / M=i+8) ---
    float* part = partials + (size_t)kc * YSIZE;
    const int mofs = (lane >> 4) * 8;
    const int fc   = lane & 15;
#pragma unroll
    for (int i = 0; i < 8; ++i) {
        const int row = rowg0 + mofs + i;
        const size_t b = (size_t)(tau * NN + row) * FF + fc;
        part[b]               = acc00[i];
        part[b + 16]          = acc01[i];
        part[b + Y2NF]        = acc10[i];
        part[b + Y2NF + 16]   = acc11[i];
    }
}

// ---------------------------------------------------------------------------
// Kernel 3: fixed-order reduction over k-chunks + (alpha*b2 + bias)
// ---------------------------------------------------------------------------
__global__ void reduce_y(const float* __restrict__ partials, const float* __restrict__ alpha_p,
                         const float* __restrict__ b2, const float* __restrict__ bias,
                         float* __restrict__ Y) {
    const int i = blockIdx.x * 256 + threadIdx.x;   // YSIZE total
    float s = 0.0f;
#pragma unroll
    for (int c = 0; c < KCH; ++c) s += partials[(size_t)c * YSIZE + i];
    const int f = i & 31;
    Y[i] = s + alpha_p[0] * b2[f] + bias[f];
}

// ---------------------------------------------------------------------------
extern "C" void kernel_launch(void* const* d_in, const int* in_sizes, int n_in,
                              void* d_out, int out_size, void* d_ws, size_t ws_size,
                              hipStream_t stream) {
    const float* X     = (const float*)d_in[0];
    const float* E0    = (const float*)d_in[1];
    const float* E1    = (const float*)d_in[2];
    const float* alpha = (const float*)d_in[3];
    const float* W1    = (const float*)d_in[4];
    const float* b1    = (const float*)d_in[5];
    const float* W2    = (const float*)d_in[6];
    const float* b2    = (const float*)d_in[7];
    const float* bias  = (const float*)d_in[8];
    float*       out   = (float*)d_out;

    float* CT   = (float*)d_ws;                 // 1 MB
    float* part = CT + CT_FLOATS;               // 8 * 4 MB partials

    prep_ct<<<NN / 64, 256, 0, stream>>>(X, W1, b1, W2, alpha, CT);
    gemm_main<<<dim3(NN / 128, KCH, 2), 256, 0, stream>>>(E0, E1, CT, out, part);
    reduce_y<<<YSIZE / 256, 256, 0, stream>>>(part, alpha, b2, bias, out);
}